// NeuralCube_46084999086179
// MI455X (gfx1250) — compile-verified
//
#include <hip/hip_runtime.h>
#include <math.h>

typedef float v2f __attribute__((ext_vector_type(2)));
typedef float v8f __attribute__((ext_vector_type(8)));

#define BATCH   256
#define IN_DIM  512
#define NN      13824   // 24^3
#define NNP     (NN + 2)  // padded h row: slot NN is the always-zero pad
#define OUT_DIM 1000
#define STEPS   30
#define SPLITK  8
#define KC      (NN / SPLITK)        // 1728
#define TN_OUT  ((OUT_DIM + 15) / 16) // 63
#define PCOLS   (TN_OUT * 16)        // 1008 padded partial columns

// ---------------------------------------------------------------------------
// Transpose W_local [N,27] -> [27,N] and neighbor_indices likewise, so the
// step kernel's per-k loads are lane-coalesced.
// ---------------------------------------------------------------------------
__global__ __launch_bounds__(256) void prep_kernel(
    const float* __restrict__ wl, const int* __restrict__ nbr,
    float* __restrict__ wlT, int* __restrict__ nbrT)
{
    int t = blockIdx.x * 256 + threadIdx.x;
    if (t >= 27 * NN) return;
    int k = t / NN;
    int n = t - k * NN;
    wlT[t]  = wl[n * 27 + k];
    nbrT[t] = nbr[n * 27 + k];
}

// Zero the pad slots of both h buffers (re-run every launch: ws is poisoned).
__global__ __launch_bounds__(256) void pad_kernel(float* hA, float* hB)
{
    int t = threadIdx.x;  // single block of 256 = BATCH
    hA[(size_t)t * NNP + NN]     = 0.0f;
    hA[(size_t)t * NNP + NN + 1] = 0.0f;
    hB[(size_t)t * NNP + NN]     = 0.0f;
    hB[(size_t)t * NNP + NN + 1] = 0.0f;
}

// ---------------------------------------------------------------------------
// GEMM 1: x_proj[b,n] = sum_k x[b,k] * W_in[n,k] + bias[n]
// One wave32 per 16x16 tile, K-loop via v_wmma_f32_16x16x4_f32 (full f32).
// A-frag (16x4 f32): lanes 0-15: M=lane, VGPR0/1=K0/K1; lanes 16-31: K2/K3.
// B-frag (4x16 f32): symmetric. C/D: 8 VGPRs, M=i+8*(lane>=16), N=lane&15.
// ---------------------------------------------------------------------------
__global__ __launch_bounds__(128) void gemm_in_kernel(
    const float* __restrict__ x, const float* __restrict__ w,
    const float* __restrict__ bias, float* __restrict__ out)
{
    const int TILES_N = NN / 16;                      // 864
    int wave = blockIdx.x * (blockDim.x >> 5) + (threadIdx.x >> 5);
    int lane = threadIdx.x & 31;
    int tm = wave / TILES_N;                          // 0..15
    int tn = wave % TILES_N;
    int half = lane >> 4;
    int l16  = lane & 15;
    int row = tm * 16 + l16;                          // batch row (A, M)
    int col = tn * 16 + l16;                          // neuron col (B, N)

    const float* aptr = x + (size_t)row * IN_DIM + half * 2;
    const float* bptr = w + (size_t)col * IN_DIM + half * 2;

    v8f c;
    float bb = bias[col];
#pragma unroll
    for (int i = 0; i < 8; ++i) c[i] = bb;

#pragma unroll 8
    for (int k = 0; k < IN_DIM; k += 4) {
        v2f a = *(const v2f*)(aptr + k);
        v2f b = *(const v2f*)(bptr + k);
        c = __builtin_amdgcn_wmma_f32_16x16x4_f32(
                false, a, false, b, (short)0, c, false, false);
    }

    int mbase = tm * 16 + half * 8;
#pragma unroll
    for (int i = 0; i < 8; ++i)
        out[(size_t)(mbase + i) * NN + col] = c[i];
}

// ---------------------------------------------------------------------------
// Recurrent stencil step: h'[b,n] = tanh(xproj[b,n] + sum_k h[b,nbr[n][k]]*W[n][k])
// Per-neuron weights -> VALU FMA. One thread handles 4 batch rows of one
// neuron: the 27 index/weight loads amortize over 4 rows, and the zero-padded
// h rows (slot NN == 0) remove the per-tap bounds select entirely.
// ---------------------------------------------------------------------------
__global__ __launch_bounds__(256) void step_kernel(
    const float* __restrict__ xproj, const float* __restrict__ hprev,
    float* __restrict__ hnext, const float* __restrict__ wlT,
    const int* __restrict__ nbrT, int first)
{
    int t = blockIdx.x * 256 + threadIdx.x;           // over (BATCH/4)*NN
    if (t >= (BATCH / 4) * NN) return;
    int bg = t / NN;
    int n  = t - bg * NN;
    int b0 = bg * 4;

    float acc0 = xproj[(size_t)(b0 + 0) * NN + n];
    float acc1 = xproj[(size_t)(b0 + 1) * NN + n];
    float acc2 = xproj[(size_t)(b0 + 2) * NN + n];
    float acc3 = xproj[(size_t)(b0 + 3) * NN + n];

    if (!first) {
        const float* h0 = hprev + (size_t)(b0 + 0) * NNP;
        const float* h1 = hprev + (size_t)(b0 + 1) * NNP;
        const float* h2 = hprev + (size_t)(b0 + 2) * NNP;
        const float* h3 = hprev + (size_t)(b0 + 3) * NNP;
#pragma unroll
        for (int k = 0; k < 27; ++k) {
            int   j = nbrT[k * NN + n];               // j==NN -> pad slot (0)
            float w = wlT[k * NN + n];
            acc0 = fmaf(h0[j], w, acc0);
            acc1 = fmaf(h1[j], w, acc1);
            acc2 = fmaf(h2[j], w, acc2);
            acc3 = fmaf(h3[j], w, acc3);
        }
    }

    hnext[(size_t)(b0 + 0) * NNP + n] = tanhf(acc0);
    hnext[(size_t)(b0 + 1) * NNP + n] = tanhf(acc1);
    hnext[(size_t)(b0 + 2) * NNP + n] = tanhf(acc2);
    hnext[(size_t)(b0 + 3) * NNP + n] = tanhf(acc3);
}

// ---------------------------------------------------------------------------
// GEMM 2 (split-K): partial[s][b][pcol] = sum_{k in slice s} h[b,k]*W_out[o,k]
// K split 8 ways -> 8064 waves for latency hiding. B-column loads clamped
// (garbage columns land in unstored/ignored partial columns); EXEC stays
// all-ones across every WMMA. Deterministic reduce adds slices + bias.
// ---------------------------------------------------------------------------
__global__ __launch_bounds__(128) void gemm_out_kernel(
    const float* __restrict__ h, const float* __restrict__ w,
    float* __restrict__ partial)
{
    int wave = blockIdx.x * (blockDim.x >> 5) + (threadIdx.x >> 5);
    int lane = threadIdx.x & 31;
    int ks  = wave / (16 * TN_OUT);                   // K-slice 0..7
    int tmn = wave % (16 * TN_OUT);
    int tm  = tmn / TN_OUT;                           // 0..15
    int tn  = tmn % TN_OUT;                           // 0..62
    int half = lane >> 4;
    int l16  = lane & 15;
    int row  = tm * 16 + l16;                         // batch row
    int col  = tn * 16 + l16;                         // output col (may be >=1000)
    int colc = (col < OUT_DIM) ? col : (OUT_DIM - 1); // clamped for loads

    const float* aptr = h + (size_t)row  * NNP + ks * KC + half * 2;
    const float* bptr = w + (size_t)colc * NN  + ks * KC + half * 2;

    v8f c;
#pragma unroll
    for (int i = 0; i < 8; ++i) c[i] = 0.0f;

#pragma unroll 8
    for (int k = 0; k < KC; k += 4) {
        v2f a = *(const v2f*)(aptr + k);
        v2f b = *(const v2f*)(bptr + k);
        c = __builtin_amdgcn_wmma_f32_16x16x4_f32(
                false, a, false, b, (short)0, c, false, false);
    }

    int mbase = tm * 16 + half * 8;
#pragma unroll
    for (int i = 0; i < 8; ++i)
        partial[((size_t)ks * BATCH + (mbase + i)) * PCOLS + col] = c[i];
}

__global__ __launch_bounds__(256) void reduce_kernel(
    const float* __restrict__ partial, const float* __restrict__ bias,
    float* __restrict__ out)
{
    int t = blockIdx.x * 256 + threadIdx.x;           // over BATCH*OUT_DIM
    if (t >= BATCH * OUT_DIM) return;
    int b = t / OUT_DIM;
    int o = t - b * OUT_DIM;
    float acc = bias[o];
#pragma unroll
    for (int s = 0; s < SPLITK; ++s)
        acc += partial[((size_t)s * BATCH + b) * PCOLS + o];
    out[t] = acc;
}

// ---------------------------------------------------------------------------
extern "C" void kernel_launch(void* const* d_in, const int* in_sizes, int n_in,
                              void* d_out, int out_size, void* d_ws, size_t ws_size,
                              hipStream_t stream)
{
    const float* x   = (const float*)d_in[0];   // [256,512]
    const float* Wi  = (const float*)d_in[1];   // [13824,512]
    const float* bi  = (const float*)d_in[2];   // [13824]
    const float* Wl  = (const float*)d_in[3];   // [13824,27]
    const float* Wo  = (const float*)d_in[4];   // [1000,13824]
    const float* bo  = (const float*)d_in[5];   // [1000]
    const int*   nbr = (const int*)d_in[6];     // [13824,27]
    float* out = (float*)d_out;                 // [256,1000]

    // Workspace layout (floats): xproj | hA | hB | wlT | nbrT | partial
    float* ws      = (float*)d_ws;
    float* xproj   = ws;
    float* hA      = xproj + (size_t)BATCH * NN;
    float* hB      = hA    + (size_t)BATCH * NNP;
    float* wlT     = hB    + (size_t)BATCH * NNP;
    int*   nbrT    = (int*)(wlT + (size_t)27 * NN);
    float* partial = (float*)(nbrT + (size_t)27 * NN);

    // 1) transpose per-neuron tables; zero h pad slots
    prep_kernel<<<(27 * NN + 255) / 256, 256, 0, stream>>>(Wl, nbr, wlT, nbrT);
    pad_kernel<<<1, 256, 0, stream>>>(hA, hB);

    // 2) input projection GEMM: 16 * 864 tiles, 4 waves/block
    {
        int tiles = 16 * (NN / 16);
        gemm_in_kernel<<<tiles / 4, 128, 0, stream>>>(x, Wi, bi, xproj);
    }

    // 3) 30 recurrent stencil steps, ping-pong hA/hB (step 0 takes h=0)
    {
        int blocks = ((BATCH / 4) * NN + 255) / 256;  // 3456
        for (int s = 0; s < STEPS; ++s) {
            const float* src = (s & 1) ? hA : hB;     // unused when s==0
            float*       dst = (s & 1) ? hB : hA;
            step_kernel<<<blocks, 256, 0, stream>>>(xproj, src, dst, wlT, nbrT,
                                                    (s == 0) ? 1 : 0);
        }
    }

    // 4) output GEMM, split-K=8 (final h is in hB: step 29 wrote hB)
    {
        int waves = SPLITK * 16 * TN_OUT;             // 8064
        gemm_out_kernel<<<waves / 4, 128, 0, stream>>>(hB, Wo, partial);
        reduce_kernel<<<(BATCH * OUT_DIM + 255) / 256, 256, 0, stream>>>(
            partial, bo, out);
    }
}